// SoftSubSampler_1726576854732
// MI455X (gfx1250) — compile-verified
//
#include <hip/hip_runtime.h>

// SoftSubSampler for MI455X (gfx1250, wave32).
// One 1024-thread workgroup per row (B=64). Row state w (131072 f32) lives
// 52 floats/thread in VGPRs + 76 floats/thread in LDS (304KB of 320KB/WGP);
// the split keeps each wave under the ~128-VGPR budget needed for the
// mandatory 8 waves/SIMD of a 1024-thread workgroup (no scratch spills).
// All 16 softmax rounds run in-kernel with block reductions; every f32 sum
// reduction stage uses V_WMMA_F32_16X16X4_F32 (A=data, B=ones =>
// D[m][n] = x_m + x_{m+16}) for both the lane stage and the wave-partial
// stage. Working set (~130MB) stays resident in the 192MB L2; khot accumulates
// via per-thread-owned global RMW (deterministic, no atomics).
// Transcendentals use raw v_log_f32/v_exp_f32 (inputs are pre-clamped to the
// normal range, so the __ocml denorm/compensation fixups are dead weight).

#define N_ELEM   131072
#define NT       1024
#define F4_ROW   (N_ELEM / 4)        // 32768 float4 per row
#define M_TOTAL  (F4_ROW / NT)       // 32 float4 per thread
#define M_REG    13                  // 52 floats/thread in VGPRs
#define M_LDS    (M_TOTAL - M_REG)   // 76 floats/thread in LDS (304KB/block)
#define K_SEL    16
#define INV_TEMP 10.0f
#define TINY_F   1.1754943508222875e-38f
#define LN2F     0.6931471805599453f
#define LOG2EF   1.4426950408889634f
#define EXPK1    (INV_TEMP * LOG2EF)   // exp((x-M)/T) = exp2(x*EXPK1 - M*EXPK1)

typedef float v2f __attribute__((ext_vector_type(2)));
typedef float v8f __attribute__((ext_vector_type(8)));

// --- raw-hardware transcendentals (valid for normal-range inputs) ---
__device__ __forceinline__ float fast_log(float x) {   // natural log, x normal
#if defined(__HIP_DEVICE_COMPILE__)
  return __builtin_amdgcn_logf(x) * LN2F;              // v_log_f32 + v_mul
#else
  return logf(x);
#endif
}
__device__ __forceinline__ float fast_exp2(float x) {
#if defined(__HIP_DEVICE_COMPILE__)
  return __builtin_amdgcn_exp2f(x);                    // v_exp_f32
#else
  return exp2f(x);
#endif
}
__device__ __forceinline__ float fast_rcp(float x) {
#if defined(__HIP_DEVICE_COMPILE__)
  return __builtin_amdgcn_rcpf(x);                     // v_rcp_f32
#else
  return 1.0f / x;
#endif
}

__device__ __forceinline__ float wave_reduce_max(float v) {
#pragma unroll
  for (int off = 16; off > 0; off >>= 1)
    v = fmaxf(v, __shfl_xor(v, off, 32));
  return v;
}

__device__ __forceinline__ int wave_reduce_sum_i(int v) {
#pragma unroll
  for (int off = 16; off > 0; off >>= 1)
    v += __shfl_xor(v, off, 32);
  return v;
}

// Full-precision f32 wave sum. WMMA stage: A(16x4)=lane partials in cols {0,2},
// B(4x16)=ones, C=0 -> D[m][n] = x_m + x_{m+16}. Summing the 8 D regs plus one
// swapx16 shuffle-add yields the full 32-lane sum in every lane.
__device__ __forceinline__ float wave_reduce_sum_f(float v) {
#if defined(__gfx1250__) && __has_builtin(__builtin_amdgcn_wmma_f32_16x16x4_f32)
  v2f a; a[0] = v;    a[1] = 0.0f;
  v2f b; b[0] = 1.0f; b[1] = 1.0f;
  v8f c = {};
  v8f d = __builtin_amdgcn_wmma_f32_16x16x4_f32(
      false, a, false, b, (short)0, c, false, false);
  float t = ((d[0] + d[1]) + (d[2] + d[3])) + ((d[4] + d[5]) + (d[6] + d[7]));
  t += __shfl_xor(t, 16, 32);
  return t;
#else
#pragma unroll
  for (int off = 16; off > 0; off >>= 1)
    v += __shfl_xor(v, off, 32);
  return v;
#endif
}

// Block reductions: lane stage -> one value per wave in LDS -> every thread
// reloads red[lane] and repeats the wave stage. All waves compute the
// identical fixed-order result (deterministic).
__device__ __forceinline__ float block_reduce_max(float v, float* red,
                                                  int lane, int wid) {
  v = wave_reduce_max(v);
  __syncthreads();
  if (lane == 0) red[wid] = v;
  __syncthreads();
  return wave_reduce_max(red[lane]);
}

__device__ __forceinline__ float block_reduce_sum_f(float v, float* red,
                                                    int lane, int wid) {
  v = wave_reduce_sum_f(v);
  __syncthreads();
  if (lane == 0) red[wid] = v;
  __syncthreads();
  return wave_reduce_sum_f(red[lane]);
}

__device__ __forceinline__ int block_reduce_sum_i(int v, int* red,
                                                  int lane, int wid) {
  v = wave_reduce_sum_i(v);
  __syncthreads();
  if (lane == 0) red[wid] = v;
  __syncthreads();
  return wave_reduce_sum_i(red[lane]);
}

__device__ __forceinline__ float gumbelize(float q) {
  q = fmaxf(q, TINY_F);            // clamp_probs lower bound (upper is 1.0f in f32)
  return -fast_log(-fast_log(q));  // -log(u) in (0, 88.7], always normal
}

__device__ __forceinline__ float flt_max4(float4 x, float lm) {
  return fmaxf(fmaxf(lm, fmaxf(x.x, x.y)), fmaxf(x.z, x.w));
}
__device__ __forceinline__ float flt_max4_below(float4 x, float thr, float lm) {
  if (x.x < thr) lm = fmaxf(lm, x.x);
  if (x.y < thr) lm = fmaxf(lm, x.y);
  if (x.z < thr) lm = fmaxf(lm, x.z);
  if (x.w < thr) lm = fmaxf(lm, x.w);
  return lm;
}
__device__ __forceinline__ int count_eq4(float4 x, float v) {
  return (int)(x.x == v) + (int)(x.y == v) + (int)(x.z == v) + (int)(x.w == v);
}
// sum of exp2(x*EXPK1 + k0) over 4 components (k0 = -M*EXPK1, uniform)
__device__ __forceinline__ float expsum4(float4 x, float k0) {
  return fast_exp2(fmaf(x.x, EXPK1, k0)) + fast_exp2(fmaf(x.y, EXPK1, k0)) +
         fast_exp2(fmaf(x.z, EXPK1, k0)) + fast_exp2(fmaf(x.w, EXPK1, k0));
}
__device__ __forceinline__ float4 softmax4(float4 x, float k0, float rS) {
  float4 p;
  p.x = fast_exp2(fmaf(x.x, EXPK1, k0)) * rS;
  p.y = fast_exp2(fmaf(x.y, EXPK1, k0)) * rS;
  p.z = fast_exp2(fmaf(x.z, EXPK1, k0)) * rS;
  p.w = fast_exp2(fmaf(x.w, EXPK1, k0)) * rS;
  return p;
}
__device__ __forceinline__ float4 suppress4(float4 x, float4 p) {
  x.x += fast_log(fmaxf(1.0f - p.x, TINY_F));
  x.y += fast_log(fmaxf(1.0f - p.y, TINY_F));
  x.z += fast_log(fmaxf(1.0f - p.z, TINY_F));
  x.w += fast_log(fmaxf(1.0f - p.w, TINY_F));
  return x;
}
__device__ __forceinline__ float4 add4(float4 a, float4 b) {
  float4 r; r.x = a.x + b.x; r.y = a.y + b.y; r.z = a.z + b.z; r.w = a.w + b.w;
  return r;
}
__device__ __forceinline__ float4 ge4(float4 x, float t) {
  float4 r;
  r.x = (x.x >= t) ? 1.0f : 0.0f;
  r.y = (x.y >= t) ? 1.0f : 0.0f;
  r.z = (x.z >= t) ? 1.0f : 0.0f;
  r.w = (x.w >= t) ? 1.0f : 0.0f;
  return r;
}
__device__ __forceinline__ float4 gum4(float4 w, float4 u) {
  w.x += gumbelize(u.x); w.y += gumbelize(u.y);
  w.z += gumbelize(u.z); w.w += gumbelize(u.w);
  return w;
}

__global__ __launch_bounds__(NT, 1)
void soft_subsampler_kernel(const float* __restrict__ logits,
                            const float* __restrict__ u,
                            float* __restrict__ out, int B) {
  __shared__ float4 wl[M_LDS * NT];   // 304 KB row state (LDS half)
  __shared__ float  redf[32];
  __shared__ int    redi[32];

  const int row  = blockIdx.x;
  const int tid  = threadIdx.x;
  const int lane = tid & 31;
  const int wid  = tid >> 5;

  const float4* lg4 = (const float4*)logits + (size_t)row * F4_ROW;
  const float4* uu4 = (const float4*)u      + (size_t)row * F4_ROW;
  float4* outd4 = (float4*)out + (size_t)row * F4_ROW;
  float4* outc4 = (float4*)out + ((size_t)B + (size_t)row) * F4_ROW;

  float4 wr[M_REG];   // register half of the row state

  // ---- load logits (single pass from memory) ----
#pragma unroll
  for (int m = 0; m < M_REG; ++m) wr[m] = lg4[tid + m * NT];
#pragma unroll
  for (int m = 0; m < M_LDS; ++m) wl[m * NT + tid] = lg4[tid + (M_REG + m) * NT];

  // ---- discrete top-K: iterative filtered max + multiplicity count ----
  // thr ends as the K-th largest logit (tie semantics match lax.top_k).
  float thr = __builtin_inff();
  int found = 0;
  for (int it = 0; it < K_SEL; ++it) {
    const bool active = (found < K_SEL);   // block-uniform
    float lm = -__builtin_inff();
    if (active) {
#pragma unroll
      for (int m = 0; m < M_REG; ++m) lm = flt_max4_below(wr[m], thr, lm);
#pragma unroll
      for (int m = 0; m < M_LDS; ++m) lm = flt_max4_below(wl[m * NT + tid], thr, lm);
    }
    const float mx = block_reduce_max(lm, redf, lane, wid);

    int lc = 0;
    if (active) {
#pragma unroll
      for (int m = 0; m < M_REG; ++m) lc += count_eq4(wr[m], mx);
#pragma unroll
      for (int m = 0; m < M_LDS; ++m) lc += count_eq4(wl[m * NT + tid], mx);
    }
    const int cnt = block_reduce_sum_i(lc, redi, lane, wid);

    if (active) { thr = mx; found += cnt; }
  }

  // ---- dsamples = (logits >= thr); zero-init csamples for RMW accumulation ----
  const float4 zero4 = {0.0f, 0.0f, 0.0f, 0.0f};
#pragma unroll
  for (int m = 0; m < M_REG; ++m) {
    outd4[tid + m * NT] = ge4(wr[m], thr);
    outc4[tid + m * NT] = zero4;
  }
#pragma unroll
  for (int m = 0; m < M_LDS; ++m) {
    outd4[tid + (M_REG + m) * NT] = ge4(wl[m * NT + tid], thr);
    outc4[tid + (M_REG + m) * NT] = zero4;
  }

  // ---- inject Gumbel noise: w = logits + (-log(-log(clamp(u)))) ----
#pragma unroll
  for (int m = 0; m < M_REG; ++m) wr[m] = gum4(wr[m], uu4[tid + m * NT]);
#pragma unroll
  for (int m = 0; m < M_LDS; ++m) {
    float4 f = gum4(wl[m * NT + tid], uu4[tid + (M_REG + m) * NT]);
    wl[m * NT + tid] = f;
  }

  // ---- 16 relaxed-softmax rounds with loop-carried suppression ----
  for (int it = 0; it < K_SEL; ++it) {
    // row max of w
    float lm = -__builtin_inff();
#pragma unroll
    for (int m = 0; m < M_REG; ++m) lm = flt_max4(wr[m], lm);
#pragma unroll
    for (int m = 0; m < M_LDS; ++m) lm = flt_max4(wl[m * NT + tid], lm);
    const float M  = block_reduce_max(lm, redf, lane, wid);
    const float k0 = -M * EXPK1;   // exp((x-M)/T) = exp2(x*EXPK1 + k0)

    // row sum of exp((w - M)/T)   [WMMA-assisted, both reduction stages]
    float ls = 0.0f;
#pragma unroll
    for (int m = 0; m < M_REG; ++m) ls += expsum4(wr[m], k0);
#pragma unroll
    for (int m = 0; m < M_LDS; ++m) ls += expsum4(wl[m * NT + tid], k0);
    const float S  = block_reduce_sum_f(ls, redf, lane, wid);
    const float rS = fast_rcp(S);

    // khot += softmax; w += log(clamp(1 - softmax))
#pragma unroll
    for (int m = 0; m < M_REG; ++m) {
      float4 x = wr[m];
      float4 p = softmax4(x, k0, rS);
      outc4[tid + m * NT] = add4(outc4[tid + m * NT], p);
      wr[m] = suppress4(x, p);
    }
#pragma unroll
    for (int m = 0; m < M_LDS; ++m) {
      float4 x = wl[m * NT + tid];
      float4 p = softmax4(x, k0, rS);
      const int g = tid + (M_REG + m) * NT;
      outc4[g] = add4(outc4[g], p);
      wl[m * NT + tid] = suppress4(x, p);
    }
  }
}

extern "C" void kernel_launch(void* const* d_in, const int* in_sizes, int n_in,
                              void* d_out, int out_size, void* d_ws, size_t ws_size,
                              hipStream_t stream) {
  (void)n_in; (void)out_size; (void)d_ws; (void)ws_size;
  const float* logits = (const float*)d_in[0];
  const float* u      = (const float*)d_in[1];
  float* out = (float*)d_out;
  const int B = in_sizes[0] / N_ELEM;   // 64
  soft_subsampler_kernel<<<dim3(B), dim3(NT), 0, stream>>>(logits, u, out, B);
}